// GATConv_2216203124984
// MI455X (gfx1250) — compile-verified
//
#include <hip/hip_runtime.h>

// ---------------- problem constants (match reference) ----------------
constexpr int N_NODES = 50000;
constexpr int N_EDGES = 1600000;
constexpr int IN_F    = 256;
constexpr int N_HEADS = 2;
constexpr int OUT_F   = 64;
constexpr int HF      = N_HEADS * OUT_F;   // 128 = flattened per-node output width
constexpr float NEG_SLOPE = 0.2f;

typedef __attribute__((ext_vector_type(2))) float v2f;
typedef __attribute__((ext_vector_type(8))) float v8f;

// monotone float -> uint encoding so integer atomicMax == float max
__device__ __forceinline__ unsigned enc_f32(float x) {
  unsigned b = __float_as_uint(x);
  return (b & 0x80000000u) ? ~b : (b | 0x80000000u);
}
__device__ __forceinline__ float dec_f32(unsigned u) {
  unsigned b = (u & 0x80000000u) ? (u & 0x7FFFFFFFu) : ~u;
  return __uint_as_float(b);
}

// ---------------------------------------------------------------------
// Kernel 1: ft[N,128] = feat[N,256] @ W[256,128] with fp32 WMMA.
// Block = 256 threads (8 waves). Block handles 16 rows; wave w handles
// output columns [16w, 16w+16). K-loop in steps of 4 (V_WMMA_F32_16X16X4_F32).
// A tile staged in LDS with padded stride (260) for conflict-free reads.
// ---------------------------------------------------------------------
#define LDS_STRIDE 260
__global__ void __launch_bounds__(256) gat_proj_wmma(
    const float* __restrict__ feat, const float* __restrict__ W,
    float* __restrict__ ft) {
  __shared__ float tileA[16 * LDS_STRIDE];

  const int rowBase = blockIdx.x * 16;
  const int t = threadIdx.x;

  // stage 16x256 fp32 A-tile: 1024 float4 loads across 256 threads
#pragma unroll
  for (int i = 0; i < 4; ++i) {
    int vid = t + i * 256;       // float4 index 0..1023
    int row = vid >> 6;          // vid / 64
    int c4  = vid & 63;
    float4 v = *(const float4*)(feat + (size_t)(rowBase + row) * IN_F + c4 * 4);
    *(float4*)(&tileA[row * LDS_STRIDE + c4 * 4]) = v;
  }
  __syncthreads();

  const int wave = t >> 5;       // 0..7 -> column tile
  const int lane = t & 31;
  const int m16  = lane & 15;
  const int hi   = lane >> 4;    // half-wave selector (K pairs {0,1} vs {2,3})
  const int colBase = wave * 16;

  v8f c = {};
  for (int k = 0; k < IN_F; k += 4) {
    // A 16x4: lane(0-15) row=m16 holds K = k+{0,1}; lane(16-31) K = k+{2,3}
    v2f a = *(const v2f*)(&tileA[m16 * LDS_STRIDE + k + 2 * hi]);
    // B 4x16: row striped across lanes; VGPR0 = K {k, k+2}, VGPR1 = K {k+1, k+3}
    v2f b;
    b.x = W[(size_t)(k + 2 * hi)     * HF + colBase + m16];
    b.y = W[(size_t)(k + 2 * hi + 1) * HF + colBase + m16];
    c = __builtin_amdgcn_wmma_f32_16x16x4_f32(false, a, false, b,
                                              (short)0, c, false, false);
  }

  // D 16x16 f32: VGPR v -> M = v + 8*hi, N = m16
#pragma unroll
  for (int v = 0; v < 8; ++v) {
    int row = rowBase + v + 8 * hi;
    ft[(size_t)row * HF + colBase + m16] = c[v];
  }
}

// ---------------------------------------------------------------------
// Kernel 2: el/er[n,h] = dot(ft[n,h,:], attn_{l,r}[h,:])  (wave per (n,h))
// ---------------------------------------------------------------------
__global__ void __launch_bounds__(256) gat_el_er(
    const float* __restrict__ ft, const float* __restrict__ al,
    const float* __restrict__ ar, float* __restrict__ el,
    float* __restrict__ er) {
  int gw   = (blockIdx.x * 256 + threadIdx.x) >> 5;   // global wave id
  int lane = threadIdx.x & 31;
  int n = gw >> 1, h = gw & 1;
  if (n >= N_NODES) return;

  const float* row = ft + (size_t)n * HF + h * OUT_F;
  const float* wl  = al + h * OUT_F;
  const float* wr  = ar + h * OUT_F;
  float x0 = row[lane], x1 = row[lane + 32];
  float sl = x0 * wl[lane] + x1 * wl[lane + 32];
  float sr = x0 * wr[lane] + x1 * wr[lane + 32];
#pragma unroll
  for (int o = 16; o > 0; o >>= 1) {
    sl += __shfl_down(sl, o, 32);
    sr += __shfl_down(sr, o, 32);
  }
  if (lane == 0) { el[n * 2 + h] = sl; er[n * 2 + h] = sr; }
}

// ---------------------------------------------------------------------
// Kernel 3: per-edge leaky-relu logits + segment max (encoded u32 atomicMax)
// ---------------------------------------------------------------------
__global__ void __launch_bounds__(256) gat_edge_logits(
    const int* __restrict__ src, const int* __restrict__ dst,
    const float* __restrict__ el, const float* __restrict__ er,
    float* __restrict__ ebuf, unsigned* __restrict__ menc) {
  int e = blockIdx.x * 256 + threadIdx.x;
  if (e >= N_EDGES) return;
  int sN = src[e], dN = dst[e];
#pragma unroll
  for (int h = 0; h < 2; ++h) {
    float v = el[sN * 2 + h] + er[dN * 2 + h];
    v = v > 0.0f ? v : NEG_SLOPE * v;
    ebuf[2 * (size_t)e + h] = v;
    atomicMax(&menc[dN * 2 + h], enc_f32(v));
  }
}

// ---------------------------------------------------------------------
// Kernel 4: decode m; empty segments (enc==0 -> NaN) guarded to 0
// ---------------------------------------------------------------------
__global__ void __launch_bounds__(256) gat_finalize_m(
    const unsigned* __restrict__ menc, float* __restrict__ m) {
  int i = blockIdx.x * 256 + threadIdx.x;
  if (i >= N_NODES * 2) return;
  float v = dec_f32(menc[i]);
  unsigned b = __float_as_uint(v);
  if ((b & 0x7F800000u) == 0x7F800000u) v = 0.0f;  // non-finite -> 0
  m[i] = v;
}

// ---------------------------------------------------------------------
// Kernel 5: e_exp = exp(e - m[dst]); segment sum into s
// ---------------------------------------------------------------------
__global__ void __launch_bounds__(256) gat_edge_exp(
    const int* __restrict__ dst, const float* __restrict__ ebuf,
    const float* __restrict__ m, float* __restrict__ s) {
  int e = blockIdx.x * 256 + threadIdx.x;
  if (e >= N_EDGES) return;
  int dN = dst[e];
#pragma unroll
  for (int h = 0; h < 2; ++h) {
    float w = expf(ebuf[2 * (size_t)e + h] - m[dN * 2 + h]);
    atomicAdd(&s[dN * 2 + h], w);
  }
}

// ---------------------------------------------------------------------
// Kernel 6: wave per edge; lane l owns flat features [4l,4l+4):
// out[dst] += a_h * ft[src]   (coalesced 512B per wave, f32 atomics in L2)
// ---------------------------------------------------------------------
__global__ void __launch_bounds__(256) gat_aggregate(
    const int* __restrict__ src, const int* __restrict__ dst,
    const float* __restrict__ ebuf, const float* __restrict__ m,
    const float* __restrict__ s, const float* __restrict__ ft,
    float* __restrict__ out) {
  int tid  = blockIdx.x * 256 + threadIdx.x;
  int e    = tid >> 5;
  int lane = tid & 31;
  if (e >= N_EDGES) return;
  int sN = src[e], dN = dst[e];
  int f0 = lane * 4;
  int h  = f0 >> 6;  // 0 for features 0..63, 1 for 64..127
  float a = expf(ebuf[2 * (size_t)e + h] - m[dN * 2 + h]) /
            fmaxf(s[dN * 2 + h], 1e-20f);
  float4 v = *(const float4*)(ft + (size_t)sN * HF + f0);
  float* o = out + (size_t)dN * HF + f0;
  atomicAdd(o + 0, a * v.x);
  atomicAdd(o + 1, a * v.y);
  atomicAdd(o + 2, a * v.z);
  atomicAdd(o + 3, a * v.w);
}

// ---------------------------------------------------------------------
extern "C" void kernel_launch(void* const* d_in, const int* in_sizes, int n_in,
                              void* d_out, int out_size, void* d_ws, size_t ws_size,
                              hipStream_t stream) {
  const float* feat = (const float*)d_in[0];
  const int*   src  = (const int*)d_in[1];
  const int*   dst  = (const int*)d_in[2];
  const float* W    = (const float*)d_in[3];
  const float* al   = (const float*)d_in[4];
  const float* ar   = (const float*)d_in[5];
  float* out = (float*)d_out;

  // workspace layout (floats):
  //   ft[N*128] | el[N*2] | er[N*2] | m_enc[N*2](u32) | m[N*2] | s[N*2] | e[E*2]
  float*    ft   = (float*)d_ws;
  float*    el   = ft + (size_t)N_NODES * HF;
  float*    er   = el + (size_t)N_NODES * 2;
  unsigned* menc = (unsigned*)(er + (size_t)N_NODES * 2);
  float*    m    = (float*)(menc + (size_t)N_NODES * 2);
  float*    s    = m + (size_t)N_NODES * 2;
  float*    ebuf = s + (size_t)N_NODES * 2;

  // per-call init (graph-capture safe): accumulators must start at zero
  (void)hipMemsetAsync(out,  0, sizeof(float) * (size_t)out_size, stream);
  (void)hipMemsetAsync(menc, 0, sizeof(unsigned) * (size_t)N_NODES * 2, stream);
  (void)hipMemsetAsync(s,    0, sizeof(float) * (size_t)N_NODES * 2, stream);

  gat_proj_wmma<<<N_NODES / 16, 256, 0, stream>>>(feat, W, ft);          // 3125 blocks
  gat_el_er<<<(N_NODES * 2 * 32) / 256, 256, 0, stream>>>(ft, al, ar, el, er);
  gat_edge_logits<<<(N_EDGES + 255) / 256, 256, 0, stream>>>(src, dst, el, er, ebuf, menc);
  gat_finalize_m<<<(N_NODES * 2 + 255) / 256, 256, 0, stream>>>(menc, m);
  gat_edge_exp<<<(N_EDGES + 255) / 256, 256, 0, stream>>>(dst, ebuf, m, s);
  gat_aggregate<<<(N_EDGES * 32) / 256, 256, 0, stream>>>(src, dst, ebuf, m, s, ft, out);
}